// RelationalNetwork_42434276884668
// MI455X (gfx1250) — compile-verified
//
#include <hip/hip_runtime.h>
#include <hip/hip_bf16.h>

// Problem: B=32, N1=N2=64, D1=D2=512, DG=256, DF=256
//
// Identity: mean over the 4096 (i,j) pairs of a linear layer == the linear
// layer applied to the mean pair vector:
//   g[b] = concat(mean_i O1[b,i,:], mean_j O2[b,j,:]) @ Wg + bg
// This turns a 68.7 GFLOP batched pair-GEMM into an HBM-bound 8.4 MB
// reduction plus ~21 MFLOP of tiny GEMMs done with V_WMMA_F32_16X16X4_F32.
//
// Roofline: 8.4 MB O1/O2 stream @ 23.3 TB/s ~ 0.4 us. Weights (Wg 1MB,
// Wf1 256KB) are spread across 16 workgroups so no single WGP's load path
// serializes the pipeline (a 1-workgroup GEMM reading 1MB would cost ~2-3us
// by itself — worse than the whole streaming phase).

typedef float v2f __attribute__((ext_vector_type(2)));
typedef float v8f __attribute__((ext_vector_type(8)));

#define WMMA_F32X4(A, B, C) \
    __builtin_amdgcn_wmma_f32_16x16x4_f32(false, (A), false, (B), (short)0, (C), false, false)

// ---------------------------------------------------------------------------
// Kernel 1: m[b, 0:512]   = mean_i O1[b, i, :]
//           m[b, 512:1024]= mean_j O2[b, j, :]
// Pure HBM streaming (8.4 MB in, 128 KB out), coalesced across threads.
// Non-temporal loads: O1/O2 are dead after this pass, keep them out of L2.
// grid = (32, 4), block = 256 -> one thread per (b, column)
// ---------------------------------------------------------------------------
__global__ __launch_bounds__(256) void rn_mean_kernel(
    const float* __restrict__ O1,   // [32, 64, 512]
    const float* __restrict__ O2,   // [32, 64, 512]
    float* __restrict__ m)          // [32, 1024]
{
    const int b = blockIdx.x;                        // 0..31
    const int c = blockIdx.y * 256 + threadIdx.x;    // 0..1023
    const float* src = (c < 512)
        ? (O1 + (size_t)b * 64 * 512 + c)
        : (O2 + (size_t)b * 64 * 512 + (c - 512));
    float s = 0.0f;
#pragma unroll
    for (int i = 0; i < 64; ++i) s += __builtin_nontemporal_load(src + i * 512);
    m[b * 1024 + c] = s * (1.0f / 64.0f);
}

// ---------------------------------------------------------------------------
// Kernel 2: g = m[32,1024] @ Wg[1024,256] + bg, via V_WMMA_F32_16X16X4_F32.
// grid = 16 blocks (one per 16-column N-tile), block = 256 (8 wave32s).
// Wave w accumulates the K-chunk [128w, 128w+128) into a 32x16 partial tile
// (2 M-tiles x 1 N-tile); partials are reduced across waves in LDS.
//
// Fragment layouts per ISA 7.12.2 (wave32), hf = lane>>4, l = lane&15:
//   A (16x4):  a = {A[l, k+2hf], A[l, k+2hf+1]}       (8B-aligned float2)
//   B (4x16):  b = {B[k+2hf, c0+l], B[k+2hf+1, c0+l]}
//   C/D:       acc[r] = D[r + 8hf, c0+l]
// ---------------------------------------------------------------------------
__global__ __launch_bounds__(256) void rn_gemm1_kernel(
    const float* __restrict__ m,    // [32, 1024]
    const float* __restrict__ Wg,   // [1024, 256]
    const float* __restrict__ bg,   // [256]
    float* __restrict__ g)          // [32, 256]
{
    __shared__ float part[8 * 32 * 16];  // 16 KB: per-wave partial tiles

    const int tid  = threadIdx.x;
    const int lane = tid & 31;
    const int wv   = tid >> 5;       // wave 0..7
    const int hf   = lane >> 4;      // lane half: 0/1
    const int l    = lane & 15;

    const int c0    = blockIdx.x * 16;   // this block's 16 columns
    const int kbase = wv * 128;          // this wave's K chunk

    v8f acc0 = {};                   // rows 0-15
    v8f acc1 = {};                   // rows 16-31

#pragma unroll 4
    for (int kk = 0; kk < 128; kk += 4) {
        const int ka = kbase + kk + 2 * hf;
        v2f a0 = *(const v2f*)(m + (0  + l) * 1024 + ka);
        v2f a1 = *(const v2f*)(m + (16 + l) * 1024 + ka);
        v2f b;
        b.x = Wg[(ka + 0) * 256 + c0 + l];
        b.y = Wg[(ka + 1) * 256 + c0 + l];
        acc0 = WMMA_F32X4(a0, b, acc0);
        acc1 = WMMA_F32X4(a1, b, acc1);
    }

    // stash this wave's 32x16 partial tile
#pragma unroll
    for (int r = 0; r < 8; ++r) {
        const int row = r + 8 * hf;
        part[wv * 512 + row * 16 + l]        = acc0[r];
        part[wv * 512 + (16 + row) * 16 + l] = acc1[r];
    }
    __syncthreads();

    // cross-wave reduction: 512 outputs, 256 threads -> 2 each
#pragma unroll
    for (int t = 0; t < 2; ++t) {
        const int idx = tid + t * 256;       // 0..511
        const int row = idx >> 4;
        const int col = idx & 15;
        float s = 0.0f;
#pragma unroll
        for (int w = 0; w < 8; ++w) s += part[w * 512 + idx];
        g[row * 256 + c0 + col] = s + bg[c0 + col];
    }
}

// ---------------------------------------------------------------------------
// Kernel 3: h = g[32,256] @ Wf1[256,256] + bf1. Same scheme, K=256 split 8x.
// ---------------------------------------------------------------------------
__global__ __launch_bounds__(256) void rn_gemm2_kernel(
    const float* __restrict__ g,    // [32, 256]
    const float* __restrict__ Wf1,  // [256, 256]
    const float* __restrict__ bf1,  // [256]
    float* __restrict__ h)          // [32, 256]
{
    __shared__ float part[8 * 32 * 16];  // 16 KB

    const int tid  = threadIdx.x;
    const int lane = tid & 31;
    const int wv   = tid >> 5;
    const int hf   = lane >> 4;
    const int l    = lane & 15;

    const int c0    = blockIdx.x * 16;
    const int kbase = wv * 32;           // K=256 split into 8 chunks of 32

    v8f acc0 = {};
    v8f acc1 = {};

#pragma unroll
    for (int kk = 0; kk < 32; kk += 4) {
        const int ka = kbase + kk + 2 * hf;
        v2f a0 = *(const v2f*)(g + (0  + l) * 256 + ka);
        v2f a1 = *(const v2f*)(g + (16 + l) * 256 + ka);
        v2f b;
        b.x = Wf1[(ka + 0) * 256 + c0 + l];
        b.y = Wf1[(ka + 1) * 256 + c0 + l];
        acc0 = WMMA_F32X4(a0, b, acc0);
        acc1 = WMMA_F32X4(a1, b, acc1);
    }

#pragma unroll
    for (int r = 0; r < 8; ++r) {
        const int row = r + 8 * hf;
        part[wv * 512 + row * 16 + l]        = acc0[r];
        part[wv * 512 + (16 + row) * 16 + l] = acc1[r];
    }
    __syncthreads();

#pragma unroll
    for (int t = 0; t < 2; ++t) {
        const int idx = tid + t * 256;
        const int row = idx >> 4;
        const int col = idx & 15;
        float s = 0.0f;
#pragma unroll
        for (int w = 0; w < 8; ++w) s += part[w * 512 + idx];
        h[row * 256 + c0 + col] = s + bf1[c0 + col];
    }
}

// ---------------------------------------------------------------------------
// Kernel 4: out[b] = sigmoid(h[b,:] . Wf2 + bf2). One wave, float4 dots.
// ---------------------------------------------------------------------------
__global__ __launch_bounds__(32) void rn_final_kernel(
    const float* __restrict__ h,    // [32, 256]
    const float* __restrict__ Wf2,  // [256]
    const float* __restrict__ bf2,  // [1]
    float* __restrict__ out)        // [32]
{
    const int b = threadIdx.x;      // 0..31
    const float4* hp = (const float4*)(h + b * 256);
    const float4* wp = (const float4*)Wf2;
    float s = bf2[0];
#pragma unroll 4
    for (int k = 0; k < 64; ++k) {
        float4 hv = hp[k];
        float4 wv = wp[k];
        s += hv.x * wv.x + hv.y * wv.y + hv.z * wv.z + hv.w * wv.w;
    }
    out[b] = 1.0f / (1.0f + __expf(-s));
}

// ---------------------------------------------------------------------------
extern "C" void kernel_launch(void* const* d_in, const int* in_sizes, int n_in,
                              void* d_out, int out_size, void* d_ws, size_t ws_size,
                              hipStream_t stream) {
    const float* O1  = (const float*)d_in[0];  // [32,64,512]
    const float* O2  = (const float*)d_in[1];  // [32,64,512]
    const float* Wg  = (const float*)d_in[2];  // [1024,256]
    const float* bg  = (const float*)d_in[3];  // [256]
    const float* Wf1 = (const float*)d_in[4];  // [256,256]
    const float* bf1 = (const float*)d_in[5];  // [256]
    const float* Wf2 = (const float*)d_in[6];  // [256]
    const float* bf2 = (const float*)d_in[7];  // [1]

    char* ws = (char*)d_ws;
    float* m = (float*)(ws);                   // [32,1024] 128 KB
    float* g = (float*)(ws + 131072);          // [32, 256]  32 KB
    float* h = (float*)(ws + 163840);          // [32, 256]  32 KB

    rn_mean_kernel <<<dim3(32, 4), 256, 0, stream>>>(O1, O2, m);
    rn_gemm1_kernel<<<16, 256, 0, stream>>>(m, Wg, bg, g);
    rn_gemm2_kernel<<<16, 256, 0, stream>>>(g, Wf1, bf1, h);
    rn_final_kernel<<<1, 32, 0, stream>>>(h, Wf2, bf2, (float*)d_out);
}